// BARO_89318139887969
// MI455X (gfx1250) — compile-verified
//
#include <hip/hip_runtime.h>
#include <hip/hip_bf16.h>
#include <math.h>

// Problem constants
#define Bsz   8
#define Wsz   32
#define Nsz   400
#define Hsz   256
#define BW    256          // B*W
#define TOPK  10
#define BN_EPS 1e-5f

typedef __attribute__((ext_vector_type(2))) float v2f;
typedef __attribute__((ext_vector_type(8))) float v8f;

// ---------------------------------------------------------------------------
// Kernel A: x_readout[bw][h] = mean_n x[bw][n][h] + max_n x[bw][n][h]
// 256 blocks (one per (b,w)), 256 threads (one per h). Fully coalesced.
// ---------------------------------------------------------------------------
__global__ __launch_bounds__(256) void k_readout(const float* __restrict__ x,
                                                 float* __restrict__ xr) {
    const int bw = blockIdx.x;
    const int h  = threadIdx.x;
    const float* xp = x + (size_t)bw * Nsz * Hsz + h;
    float s  = 0.0f;
    float mx = -INFINITY;
    #pragma unroll 4
    for (int n = 0; n < Nsz; ++n) {
        float v = xp[(size_t)n * Hsz];
        s += v;
        mx = fmaxf(mx, v);
    }
    xr[bw * Hsz + h] = s * (1.0f / (float)Nsz) + mx;
}

// ---------------------------------------------------------------------------
// Kernel B: e = GELU(BN(xr @ W_embed + b_embed))   [256 x 256]
// One wave32 per 16x16 output tile, f32 WMMA 16x16x4 chain over K=256.
// 256 tiles -> 32 blocks x 8 waves.
// ---------------------------------------------------------------------------
__global__ __launch_bounds__(256) void k_embed(const float* __restrict__ A,    // xr [256,256]
                                               const float* __restrict__ Wm,   // W_embed [256,256]
                                               const float* __restrict__ bias, // b_embed [256]
                                               const float* __restrict__ gamma,
                                               const float* __restrict__ beta,
                                               const float* __restrict__ mean,
                                               const float* __restrict__ var,
                                               float* __restrict__ E) {        // e [256,256]
    const int wave = threadIdx.x >> 5;
    const int lane = threadIdx.x & 31;
    const int tile = blockIdx.x * 8 + wave;          // 0..255
    const int mt = tile >> 4;                        // 16 row tiles
    const int nt = tile & 15;                        // 16 col tiles
    const int half = lane >> 4;                      // 0: K lo pair, 1: K hi pair
    const int l    = lane & 15;

    const int m = mt * 16 + l;                       // A row this lane supplies
    const int n = nt * 16 + l;                       // B col this lane supplies

    v8f acc = {};
    for (int kk = 0; kk < Hsz; kk += 4) {
        const int ka = kk + half * 2;
        v2f a, b;
        a.x = A[m * Hsz + ka];
        a.y = A[m * Hsz + ka + 1];
        b.x = Wm[(size_t)ka * Hsz + n];
        b.y = Wm[(size_t)(ka + 1) * Hsz + n];
        acc = __builtin_amdgcn_wmma_f32_16x16x4_f32(false, a, false, b,
                                                    (short)0, acc, false, false);
    }

    // Epilogue: bias + BN(eval) + exact GELU. Column (channel) = n.
    const float sc = gamma[n] * rsqrtf(var[n] + BN_EPS);
    const float bi = bias[n];
    const float mu = mean[n];
    const float be = beta[n];
    #pragma unroll
    for (int r = 0; r < 8; ++r) {
        const int mrow = mt * 16 + r + half * 8;     // C/D layout: M = r (+8 for hi half)
        float v = acc[r] + bi;
        v = (v - mu) * sc + be;
        v = 0.5f * v * (1.0f + erff(v * 0.70710678118654752f));  // exact GELU
        E[mrow * Hsz + n] = v;
    }
}

// ---------------------------------------------------------------------------
// Kernel C: gate = sigmoid(e @ W_attend + b_attend)   [256 x 400]
// Also writes transposed gate output: out_gate[w][b][n].
// 16 x 25 = 400 tiles -> 50 blocks x 8 waves.
// ---------------------------------------------------------------------------
__global__ __launch_bounds__(256) void k_attend(const float* __restrict__ E,    // [256,256]
                                                const float* __restrict__ Wa,   // [256,400]
                                                const float* __restrict__ ba,   // [400]
                                                float* __restrict__ gate,       // [256,400]
                                                float* __restrict__ gate_t) {   // [32,8,400]
    const int wave = threadIdx.x >> 5;
    const int lane = threadIdx.x & 31;
    const int tile = blockIdx.x * 8 + wave;          // 0..399
    const int mt = tile / 25;                        // 16 row tiles
    const int nt = tile % 25;                        // 25 col tiles
    const int half = lane >> 4;
    const int l    = lane & 15;

    const int m = mt * 16 + l;
    const int n = nt * 16 + l;

    v8f acc = {};
    for (int kk = 0; kk < Hsz; kk += 4) {
        const int ka = kk + half * 2;
        v2f a, b;
        a.x = E[m * Hsz + ka];
        a.y = E[m * Hsz + ka + 1];
        b.x = Wa[(size_t)ka * Nsz + n];
        b.y = Wa[(size_t)(ka + 1) * Nsz + n];
        acc = __builtin_amdgcn_wmma_f32_16x16x4_f32(false, a, false, b,
                                                    (short)0, acc, false, false);
    }

    const float bn = ba[n];
    #pragma unroll
    for (int r = 0; r < 8; ++r) {
        const int mrow = mt * 16 + r + half * 8;     // = b*W + w
        float z = acc[r] + bn;
        float g = 1.0f / (1.0f + expf(-z));
        gate[mrow * Nsz + n] = g;
        const int bb = mrow >> 5;                    // b
        const int ww = mrow & 31;                    // w
        gate_t[(ww * Bsz + bb) * Nsz + n] = g;       // transpose (1,0,2)
    }
}

// ---------------------------------------------------------------------------
// Kernel D: per (b,w): top-10 of gate, then gated accumulation over x
//   output[bw][h]   = sum_n x*gate / N
//   selected[bw][h] = sum_{n in top10} x*gate
// 256 blocks, 256 threads. x is L2-resident from kernel A's pass.
// ---------------------------------------------------------------------------
__global__ __launch_bounds__(256) void k_gather(const float* __restrict__ x,
                                                const float* __restrict__ gate,
                                                float* __restrict__ out_mean,
                                                float* __restrict__ out_sel) {
    __shared__ float gate_s[Nsz];
    __shared__ float work[Nsz];
    __shared__ float flag[Nsz];
    __shared__ float rv[256];
    __shared__ int   ri[256];

    const int bw = blockIdx.x;
    const int t  = threadIdx.x;

    for (int n = t; n < Nsz; n += 256) {
        float g = gate[bw * Nsz + n];
        gate_s[n] = g;
        work[n]   = g;
        flag[n]   = 0.0f;
    }
    __syncthreads();

    // Top-10 via iterative argmax (tie -> lowest index, matching lax.top_k).
    for (int k = 0; k < TOPK; ++k) {
        float best = -INFINITY;
        int   bi   = 0x7fffffff;
        for (int n = t; n < Nsz; n += 256) {
            float v = work[n];
            if (v > best || (v == best && n < bi)) { best = v; bi = n; }
        }
        rv[t] = best; ri[t] = bi;
        __syncthreads();
        for (int s = 128; s > 0; s >>= 1) {
            if (t < s) {
                float v2 = rv[t + s]; int i2 = ri[t + s];
                if (v2 > rv[t] || (v2 == rv[t] && i2 < ri[t])) { rv[t] = v2; ri[t] = i2; }
            }
            __syncthreads();
        }
        if (t == 0) {
            flag[ri[0]] = 1.0f;
            work[ri[0]] = -INFINITY;
        }
        __syncthreads();
    }

    // Gated accumulation; thread t owns channel h=t -> coalesced.
    const float* xp = x + (size_t)bw * Nsz * Hsz + t;
    float acc = 0.0f, accs = 0.0f;
    #pragma unroll 4
    for (int n = 0; n < Nsz; ++n) {
        float g = gate_s[n];
        float v = xp[(size_t)n * Hsz];
        float a = v * g;
        acc  += a;
        accs += flag[n] * a;
    }
    out_mean[bw * Hsz + t] = acc * (1.0f / (float)Nsz);
    out_sel[bw * Hsz + t]  = accs;
}

// ---------------------------------------------------------------------------
extern "C" void kernel_launch(void* const* d_in, const int* in_sizes, int n_in,
                              void* d_out, int out_size, void* d_ws, size_t ws_size,
                              hipStream_t stream) {
    const float* x        = (const float*)d_in[0];
    const float* W_embed  = (const float*)d_in[1];
    const float* b_embed  = (const float*)d_in[2];
    const float* bn_gamma = (const float*)d_in[3];
    const float* bn_beta  = (const float*)d_in[4];
    const float* bn_mean  = (const float*)d_in[5];
    const float* bn_var   = (const float*)d_in[6];
    const float* W_attend = (const float*)d_in[7];
    const float* b_attend = (const float*)d_in[8];

    float* out_mean = (float*)d_out;                       // [8,32,256]
    float* out_sel  = out_mean + BW * Hsz;                 // [8,32,256]
    float* out_gate = out_sel  + BW * Hsz;                 // [32,8,400]

    float* ws   = (float*)d_ws;
    float* xr   = ws;                                      // [256,256]
    float* e    = xr + BW * Hsz;                           // [256,256]
    float* gate = e  + BW * Hsz;                           // [256,400]

    k_readout<<<BW, 256, 0, stream>>>(x, xr);
    k_embed  <<<32, 256, 0, stream>>>(xr, W_embed, b_embed, bn_gamma, bn_beta,
                                      bn_mean, bn_var, e);
    k_attend <<<50, 256, 0, stream>>>(e, W_attend, b_attend, gate, out_gate);
    k_gather <<<BW, 256, 0, stream>>>(x, gate, out_mean, out_sel);
}